// LSSViewTransformer_24816321036760
// MI455X (gfx1250) — compile-verified
//
#include <hip/hip_runtime.h>
#include <hip/hip_bf16.h>
#include <math.h>

// ---------------- problem constants ----------------
constexpr int BATCH   = 4;
constexpr int NCAM    = 6;
constexpr int BN      = BATCH * NCAM;   // 24
constexpr int CIN     = 256;
constexpr int HH      = 16;
constexpr int WW      = 44;
constexpr int PIX     = HH * WW;        // 704 pixels per (b,n)
constexpr int DBINS   = 59;
constexpr int CCH     = 80;
constexpr int OCH     = DBINS + CCH;    // 139 output channels of the GEMM
constexpr int BEVH    = 128;
constexpr int BEVW    = 128;
constexpr int CELLS   = BEVH * BEVW;    // 16384
constexpr int MTILES  = 9;              // ceil(139/16)
constexpr int PTILES  = PIX / 16;       // 44

typedef __attribute__((ext_vector_type(2))) float v2f;
typedef __attribute__((ext_vector_type(8))) float v8f;

// ---------------- kernel 0: zero the BEV accumulator ----------------
__global__ void zero_bev(float* __restrict__ p, size_t n) {
    size_t i      = (size_t)blockIdx.x * blockDim.x + threadIdx.x;
    size_t stride = (size_t)gridDim.x * blockDim.x;
    for (; i < n; i += stride) p[i] = 0.0f;
}

// ---------------- kernel 1: FP32 WMMA GEMM  x = W @ img + b ----------------
// x[bn][o][p] = sum_k W[o][k] * img[bn][k][p] + bias[o]
// One wave per 16x16 output tile; wave w of the block owns M-tile w (9 waves).
// blockIdx.x = bn (24), blockIdx.y = pixel tile (44).
//
// ISA 7.12.2 layouts (32-bit, wave32):
//   A (16x4, MxK):  lane&15 -> M row; lane>=16 holds K=2,3; VGPR0/1 -> +0/+1.
//   B (4x16,  KxN):  lane&15 -> N col; lane>=16 holds K=2,3; VGPR0/1 -> +0/+1.
//   C/D (16x16):     lane&15 -> N col; VGPR r -> M = r + 8*(lane>=16).
__global__ __launch_bounds__(32 * MTILES) void gemm_wmma(
    const float* __restrict__ img,   // (24, 256, 704)
    const float* __restrict__ Wd,    // (139, 256)
    const float* __restrict__ bias,  // (139)
    float* __restrict__ x)           // (24, 139, 704)
{
    const int wave = threadIdx.x >> 5;   // 0..8 -> M tile
    const int lane = threadIdx.x & 31;
    const int bn   = blockIdx.x;         // 0..23
    const int p0   = blockIdx.y * 16;    // pixel tile origin
    const int m0   = wave * 16;
    const int hi   = lane >> 4;          // 0: K pair {0,1}, 1: K pair {2,3}
    const int l16  = lane & 15;

    // clamp A row for the partial last M-tile (stores are masked below)
    int m = m0 + l16;
    if (m > OCH - 1) m = OCH - 1;

    const float* Arow = Wd + (size_t)m * CIN + 2 * hi;                 // + k
    const float* Bcol = img + (size_t)bn * CIN * PIX                   // + k*PIX
                            + (size_t)(2 * hi) * PIX + p0 + l16;

    v8f acc = {};
#pragma unroll 4
    for (int k = 0; k < CIN; k += 4) {
        v2f a, b;
        a.x = Arow[k];
        a.y = Arow[k + 1];
        b.x = Bcol[(size_t)k * PIX];
        b.y = Bcol[(size_t)(k + 1) * PIX];
        acc = __builtin_amdgcn_wmma_f32_16x16x4_f32(
            /*neg_a=*/false, a, /*neg_b=*/false, b,
            /*c_mod=*/(short)0, acc, /*reuse_a=*/false, /*reuse_b=*/false);
    }

    // store D (+ fused bias): row o = m0 + r + 8*hi, col p = p0 + l16
    float* xo = x + (size_t)bn * OCH * PIX + p0 + l16;
#pragma unroll
    for (int r = 0; r < 8; ++r) {
        int o = m0 + r + 8 * hi;
        if (o < OCH) xo[(size_t)o * PIX] = acc[r] + bias[o];
    }
}

// ---------------- kernel 2: softmax + outer product + scatter-add ----------------
// One block (160 threads, 5 waves) per pixel. bev layout: (B, 16384 cells, 80) —
// channel-contiguous so each (d)'s 80 atomics hit ~3 cachelines.
__global__ __launch_bounds__(160) void splat(
    const float* __restrict__ x,     // (24, 139, 704)
    const int*   __restrict__ geom,  // (4, 6, 59, 16, 44, 2)
    float*       __restrict__ bev)   // (4, 16384, 80)
{
    __shared__ float sm[OCH];        // [0..58] depth logits->probs, [59..138] context
    __shared__ int   cells[DBINS];
    __shared__ float smax, ssum;

    const int p   = blockIdx.x;      // 0..16895
    const int bn  = p / PIX;
    const int pix = p - bn * PIX;
    const int b   = bn / NCAM;
    const int t   = threadIdx.x;

    if (t < OCH)
        sm[t] = x[((size_t)bn * OCH + t) * PIX + pix];
    if (t < DBINS) {
        size_t g = (((size_t)bn * DBINS + t) * PIX + pix) * 2;
        cells[t] = geom[g] * BEVW + geom[g + 1];
    }
    __syncthreads();

    // max over the 59 depth logits (wave 0)
    if (t < 32) {
        float m = (t < DBINS) ? sm[t] : -INFINITY;
        if (t + 32 < DBINS) m = fmaxf(m, sm[t + 32]);
        for (int off = 16; off; off >>= 1) m = fmaxf(m, __shfl_xor(m, off, 32));
        if (t == 0) smax = m;
    }
    __syncthreads();
    if (t < DBINS) sm[t] = __expf(sm[t] - smax);
    __syncthreads();
    if (t < 32) {
        float s = (t < DBINS) ? sm[t] : 0.0f;
        if (t + 32 < DBINS) s += sm[t + 32];
        for (int off = 16; off; off >>= 1) s += __shfl_xor(s, off, 32);
        if (t == 0) ssum = s;
    }
    __syncthreads();

    const float inv  = 1.0f / ssum;
    float* bevb = bev + (size_t)b * CELLS * CCH;
    for (int i = t; i < DBINS * CCH; i += 160) {
        const int d = i / CCH;
        const int c = i - d * CCH;
        atomicAdd(&bevb[(size_t)cells[d] * CCH + c], sm[d] * inv * sm[DBINS + c]);
    }
}

// ---------------- kernel 3: transpose bev[b][cell][c] -> out[b][c][cell] ----------------
__global__ void transpose_out(const float* __restrict__ bev, float* __restrict__ out) {
    size_t i = (size_t)blockIdx.x * blockDim.x + threadIdx.x;
    constexpr size_t TOT = (size_t)BATCH * CCH * CELLS;
    if (i >= TOT) return;
    const int cell = (int)(i & (CELLS - 1));
    const int c    = (int)((i >> 14) % CCH);
    const int b    = (int)(i / ((size_t)CCH * CELLS));
    out[i] = bev[((size_t)b * CELLS + cell) * CCH + c];
}

// ---------------- launch ----------------
extern "C" void kernel_launch(void* const* d_in, const int* in_sizes, int n_in,
                              void* d_out, int out_size, void* d_ws, size_t ws_size,
                              hipStream_t stream) {
    const float* img  = (const float*)d_in[0];  // (4,6,256,16,44)
    // d_in[1..3] = rots/trans/intrins: unused by the reference math (geom is given)
    const float* Wd   = (const float*)d_in[4];  // (139,256)
    const float* bias = (const float*)d_in[5];  // (139,)
    const int*   geom = (const int*)d_in[6];    // (4,6,59,16,44,2)
    float*       out  = (float*)d_out;          // (4,80,128,128)

    constexpr size_t X_ELEMS   = (size_t)BN * OCH * PIX;        // 2,348,544
    constexpr size_t BEV_ELEMS = (size_t)BATCH * CELLS * CCH;   // 5,242,880
    float* xbuf = (float*)d_ws;
    float* bev  = xbuf + X_ELEMS;

    zero_bev<<<1024, 256, 0, stream>>>(bev, BEV_ELEMS);

    dim3 ggrid(BN, PTILES);                     // 24 x 44 blocks, 9 waves each
    gemm_wmma<<<ggrid, 32 * MTILES, 0, stream>>>(img, Wd, bias, xbuf);

    splat<<<BN * PIX, 160, 0, stream>>>(xbuf, geom, bev);

    transpose_out<<<(unsigned)((BEV_ELEMS + 255) / 256), 256, 0, stream>>>(bev, out);
}